// VQAvgSuperPool_50027779064363
// MI455X (gfx1250) — compile-verified
//
#include <hip/hip_runtime.h>

#define B_ 8
#define N_ 13
#define L_ 4096
#define D_ 1024
#define V_ 320
#define NCH 64
#define CHL (L_ / NCH) /* 64 l's per chunk -> 16 WMMAs per wave */

typedef float v2f __attribute__((ext_vector_type(2)));
typedef float v8f __attribute__((ext_vector_type(8)));

// ---------------------------------------------------------------------------
// Kernel 1: connected-component labeling + weight computation, one block/batch
// ---------------------------------------------------------------------------
__global__ __launch_bounds__(1024) void cc_weights_kernel(
    const int* __restrict__ lengths, const int* __restrict__ vq,
    float* __restrict__ wout) {
  extern __shared__ int sm[];
  int* lab   = sm;                   // [L_]  labels
  int* scr   = sm + L_;              // [L_]  run-min scratch / comp-size counts
  int* rs    = sm + 2 * L_;          // [L_]  run-start index
  int* bmin  = sm + 3 * L_;          // [V_+1] per-b-bucket min label
  int* flags = sm + 3 * L_ + V_ + 4; // [0]=changed, [1]=num_comp

  const int b   = blockIdx.x;
  const int len = lengths[b];
  const int tid = threadIdx.x;
  const int* vqb = vq + b * L_ * 2;

  // init labels and run-start links (valid region is the prefix [0,len))
  for (int i = tid; i < L_; i += 1024) {
    lab[i] = i;
    int r = i;
    if (i > 0 && i < len && vqb[2 * i] == vqb[2 * (i - 1)]) r = i - 1;
    rs[i] = r;
  }
  __syncthreads();
  // pointer jumping: rs[i] -> start of its run (12 rounds covers 4096)
  for (int s = 0; s < 12; ++s) {
    int v0 = rs[rs[tid]];
    int v1 = rs[rs[tid + 1024]];
    int v2 = rs[rs[tid + 2048]];
    int v3 = rs[rs[tid + 3072]];
    __syncthreads();
    rs[tid] = v0; rs[tid + 1024] = v1; rs[tid + 2048] = v2; rs[tid + 3072] = v3;
    __syncthreads();
  }

  // alternating segment-min (run segments, then b buckets) to fixed point
  while (true) {
    if (tid == 0) flags[0] = 0;
    for (int i = tid; i < L_; i += 1024) scr[i] = 0x7fffffff;
    for (int i = tid; i < V_ + 1; i += 1024) bmin[i] = 0x7fffffff;
    __syncthreads();
    for (int i = tid; i < len; i += 1024) atomicMin(&scr[rs[i]], lab[i]);
    __syncthreads();
    for (int i = tid; i < len; i += 1024) lab[i] = scr[rs[i]];
    __syncthreads();
    for (int i = tid; i < len; i += 1024) atomicMin(&bmin[vqb[2 * i + 1]], lab[i]);
    __syncthreads();
    int ch = 0;
    for (int i = tid; i < len; i += 1024) {
      int nl = bmin[vqb[2 * i + 1]];
      ch |= (nl != lab[i]);
      lab[i] = nl;
    }
    if (ch) atomicOr(&flags[0], 1);
    __syncthreads();
    int done = (flags[0] == 0);
    __syncthreads();
    if (done) break;
  }

  // component sizes and component count
  if (tid == 0) flags[1] = 0;
  for (int i = tid; i < L_; i += 1024) scr[i] = 0;
  __syncthreads();
  int myc = 0;
  for (int i = tid; i < len; i += 1024) {
    atomicAdd(&scr[lab[i]], 1);
    myc += (lab[i] == i) ? 1 : 0;
  }
  if (myc) atomicAdd(&flags[1], myc);
  __syncthreads();
  const float nc = (float)flags[1];
  float* wb = wout + b * L_;
  for (int i = tid; i < L_; i += 1024) {
    float wv = 0.0f;
    if (i < len) wv = 1.0f / fmaxf((float)scr[lab[i]] * nc, 1.0f);
    wb[i] = wv;
  }
}

// ---------------------------------------------------------------------------
// Kernel 2: weighted reduction over L via V_WMMA_F32_16X16X4_F32
//   A[m][k] = feat[l0+k][d0+m]  (documented 16x4 f32 A layout)
//   B[k][n] = w[l0+k]           (broadcast across N columns)
//   Single per-lane base pointer + compile-time-constant offsets so every
//   feat load folds into global_load_b32 ..., off offset:<imm>.
// ---------------------------------------------------------------------------
__global__ __launch_bounds__(256) void pool_wmma_kernel(
    const float* __restrict__ feat, const float* __restrict__ w,
    float* __restrict__ partial) {
  const int lane = threadIdx.x & 31;
  const int wave = threadIdx.x >> 5;
  const int ch = blockIdx.y * 8 + wave;         // 0..NCH-1
  const int b  = blockIdx.z;
  const int d0 = blockIdx.x * 16;
  const int m   = lane & 15;
  const int klo = (lane >> 4) << 1;             // lanes 0-15: K=0,1 ; 16-31: K=2,3
  const int l0 = ch * CHL;

  const float* base  = feat + (size_t)(b * N_ + (N_ - 1)) * L_ * D_
                            + (size_t)(l0 + klo) * D_ + d0 + m;
  const float* wbase = w + b * L_ + l0 + klo;

  v8f c = {};
#pragma unroll
  for (int t = 0; t < 4; ++t) {
    v2f a[4], bb[4];
#pragma unroll
    for (int u = 0; u < 4; ++u) {
      const int s4 = t * 16 + 4 * u;            // compile-time constant
      a[u].x  = base[(size_t)(s4)     * D_];
      a[u].y  = base[(size_t)(s4 + 1) * D_];
      bb[u].x = wbase[s4];
      bb[u].y = wbase[s4 + 1];
    }
#pragma unroll
    for (int u = 0; u < 4; ++u)
      c = __builtin_amdgcn_wmma_f32_16x16x4_f32(false, a[u], false, bb[u],
                                                (short)0, c, false, false);
  }
  // C 16x16 f32 layout: lane<16 -> rows 0..7 in c[0..7]; lane>=16 -> rows 8..15.
  // All columns identical (B broadcast), so lanes 0 and 16 emit the result.
  if (m == 0) {
    float* p = partial + (size_t)(b * NCH + ch) * D_ + d0 + (lane >> 4) * 8;
#pragma unroll
    for (int r = 0; r < 8; ++r) p[r] = c[r];
  }
}

// ---------------------------------------------------------------------------
// Kernel 3: deterministic reduction of the NCH chunk partials
// ---------------------------------------------------------------------------
__global__ __launch_bounds__(256) void reduce_kernel(
    const float* __restrict__ partial, float* __restrict__ out) {
  const int idx = blockIdx.x * 256 + threadIdx.x;   // 0..8191
  const int b = idx >> 10;
  const int d = idx & 1023;
  float s = 0.0f;
#pragma unroll 8
  for (int ch = 0; ch < NCH; ++ch)
    s += partial[(size_t)(b * NCH + ch) * D_ + d];
  out[idx] = s;
}

// ---------------------------------------------------------------------------
extern "C" void kernel_launch(void* const* d_in, const int* in_sizes, int n_in,
                              void* d_out, int out_size, void* d_ws, size_t ws_size,
                              hipStream_t stream) {
  const float* feat    = (const float*)d_in[0];   // (B,N,L,D) f32
  const int*   lengths = (const int*)d_in[1];     // (B,) i32
  const int*   vq      = (const int*)d_in[2];     // (B,L,2) i32
  float* out = (float*)d_out;                     // (B,D) f32

  float* wbuf    = (float*)d_ws;                  // B*L floats      (128 KB)
  float* partial = wbuf + B_ * L_;                // B*NCH*D floats  (2 MB)

  const size_t smem = (size_t)(3 * L_ + V_ + 8) * sizeof(int);
  cc_weights_kernel<<<B_, 1024, smem, stream>>>(lengths, vq, wbuf);
  pool_wmma_kernel<<<dim3(D_ / 16, NCH / 8, B_), 256, 0, stream>>>(feat, wbuf, partial);
  reduce_kernel<<<(B_ * D_) / 256, 256, 0, stream>>>(partial, out);
}